// Net_2783138808435
// MI455X (gfx1250) — compile-verified
//
#include <hip/hip_runtime.h>
#include <hip/hip_bf16.h>

typedef __attribute__((ext_vector_type(2))) float v2f;
typedef __attribute__((ext_vector_type(8))) float v8f;

#define BN_EPS 1e-5f

// ---------------------------------------------------------------------------
// Degree kernels
// ---------------------------------------------------------------------------
__global__ void init_deg_kernel(float* __restrict__ deg, int n) {
    int i = blockIdx.x * blockDim.x + threadIdx.x;
    if (i < n) deg[i] = 1.0f;  // self loop
}

__global__ void count_deg_kernel(const int* __restrict__ row, float* __restrict__ deg, int e) {
    int i = blockIdx.x * blockDim.x + threadIdx.x;
    if (i < e) atomicAdd(&deg[row[i]], 1.0f);
}

__global__ void invert_deg_kernel(float* __restrict__ deg, int n) {
    int i = blockIdx.x * blockDim.x + threadIdx.x;
    if (i < n) deg[i] = 1.0f / deg[i];
}

// ---------------------------------------------------------------------------
// Dense GCN layer (self part) via fp32 WMMA, dims as compile-time constants.
//   out[i,:] = x[i,:] @ W_top + b + inv_deg[i] * z[i,:]
//   z[i,:]   = x[i,:] @ W_bot
// W is (2*DIN, DOUT) row-major, staged once into LDS per block.
// One wave computes a 16-node x 16-col tile with V_WMMA_F32_16X16X4_F32.
// ---------------------------------------------------------------------------
template <int DIN, int DOUT>
__global__ void gcn_dense_wmma_kernel(const float* __restrict__ x,
                                      const float* __restrict__ W,
                                      const float* __restrict__ bias,
                                      const float* __restrict__ inv_deg,
                                      float* __restrict__ out,
                                      float* __restrict__ z,
                                      int nNodes) {
    __shared__ float sW[2 * DIN * DOUT];
    for (int i = threadIdx.x; i < 2 * DIN * DOUT; i += blockDim.x)
        sW[i] = W[i];
    __syncthreads();

    const int lane = threadIdx.x & 31;
    const int wave = threadIdx.x >> 5;
    const int tile = blockIdx.x * (blockDim.x >> 5) + wave;   // 16-node tile id
    const int row0 = tile * 16;
    if (row0 >= nNodes) return;   // wave-uniform: EXEC stays all-ones for WMMA

    const int m   = lane & 15;          // A-matrix row (node in tile)
    const int kk  = (lane >> 4) << 1;   // K sub-offset: 0 (lanes 0-15) or 2 (lanes 16-31)
    const int n   = lane & 15;          // B/C column within the 16-wide n-tile
    const int mhi = (lane >> 4) << 3;   // C/D: upper lanes hold rows M=r+8

    const float* xrow = x + (size_t)(row0 + m) * DIN + kk;    // A lane base
    const float* wtL  = sW + kk * DOUT + n;                   // B lane base (top)
    const float* wbL  = sW + (DIN + kk) * DOUT + n;           // B lane base (bottom)

    const bool fullTile = (row0 + 16 <= nNodes);              // wave-uniform

#pragma unroll
    for (int nt = 0; nt < DOUT; nt += 16) {
        v8f accT, accB;
        const float bval = bias[nt + n];
#pragma unroll
        for (int i = 0; i < 8; ++i) { accT[i] = bval; accB[i] = 0.0f; }

#pragma unroll 4
        for (int k0 = 0; k0 < DIN; k0 += 4) {
            const v2f a = *(const v2f*)(xrow + k0);           // global_load_b64
            v2f bt, bb;                                        // ds_load_2addr, imm offsets
            bt.x = wtL[k0 * DOUT + nt];
            bt.y = wtL[(k0 + 1) * DOUT + nt];
            bb.x = wbL[k0 * DOUT + nt];
            bb.y = wbL[(k0 + 1) * DOUT + nt];
            accT = __builtin_amdgcn_wmma_f32_16x16x4_f32(false, a, false, bt,
                                                         (short)0, accT, false, false);
            accB = __builtin_amdgcn_wmma_f32_16x16x4_f32(false, a, false, bb,
                                                         (short)0, accB, false, false);
        }

        // Epilogue: lane covers rows row0+mhi .. row0+mhi+7, column nt+n.
        float* zp        = z       + (size_t)(row0 + mhi) * DOUT + nt + n;
        float* op        = out     + (size_t)(row0 + mhi) * DOUT + nt + n;
        const float* ip  = inv_deg + row0 + mhi;

        if (fullTile) {
            // common path: unguarded, immediate-offset stores
#pragma unroll
            for (int r = 0; r < 8; ++r) {
                const float zv = accB[r];
                zp[r * DOUT] = zv;
                op[r * DOUT] = accT[r] + ip[r] * zv;
            }
        } else {
#pragma unroll
            for (int r = 0; r < 8; ++r) {
                if (row0 + mhi + r < nNodes) {
                    const float zv = accB[r];
                    zp[r * DOUT] = zv;
                    op[r * DOUT] = accT[r] + ip[r] * zv;
                }
            }
        }
    }
}

// ---------------------------------------------------------------------------
// Edge scatter: out[row,:] += inv_deg[row] * z[col,:]
// Each thread handles 4 channels of one edge (float4 gather + 4 f32 atomics).
// All node tensors (<=25.6 MB) are L2-resident on MI455X (192 MB L2).
// ---------------------------------------------------------------------------
template <int LOG2DOUT>
__global__ void edge_scatter_kernel(const int* __restrict__ row,
                                    const int* __restrict__ col,
                                    const float* __restrict__ inv_deg,
                                    const float* __restrict__ z,
                                    float* __restrict__ out,
                                    int nEdges) {
    const int gShift = LOG2DOUT - 2;               // float4 groups per edge
    const long long t = (long long)blockIdx.x * blockDim.x + threadIdx.x;
    const long long total = (long long)nEdges << gShift;
    if (t >= total) return;
    const int e = (int)(t >> gShift);
    const int g = ((int)t & ((1 << gShift) - 1)) << 2;   // channel base
    const int r = row[e];
    const int c = col[e];
    const float s = inv_deg[r];
    const float4 zv = *(const float4*)(z + (((size_t)c) << LOG2DOUT) + g);
    float* o = out + (((size_t)r) << LOG2DOUT) + g;
    atomicAdd(o + 0, s * zv.x);
    atomicAdd(o + 1, s * zv.y);
    atomicAdd(o + 2, s * zv.z);
    atomicAdd(o + 3, s * zv.w);
}

// ---------------------------------------------------------------------------
// BatchNorm (16 channels): stats + apply (+ReLU)
// ---------------------------------------------------------------------------
__global__ void zero_stats_kernel(float* __restrict__ stats) {
    if (threadIdx.x < 32) stats[threadIdx.x] = 0.0f;
}

__global__ void bn_stats_kernel(const float* __restrict__ x,
                                float* __restrict__ stats, int nNodes) {
    const int tid = blockIdx.x * blockDim.x + threadIdx.x;
    const int c = tid & 15;
    const int rstride = (gridDim.x * blockDim.x) >> 4;
    float s = 0.0f, sq = 0.0f;
    for (int r = tid >> 4; r < nNodes; r += rstride) {
        const float v = x[(size_t)r * 16 + c];
        s += v;
        sq += v * v;
    }
    // lanes L and L^16 cover the same channel -> fold halves, one atomic per pair
    s  += __shfl_xor(s, 16, 32);
    sq += __shfl_xor(sq, 16, 32);
    if ((threadIdx.x & 31) < 16) {
        atomicAdd(&stats[c], s);
        atomicAdd(&stats[16 + c], sq);
    }
}

__global__ void bn_apply_kernel(float* __restrict__ x,
                                const float* __restrict__ stats,
                                const float* __restrict__ gamma,
                                const float* __restrict__ beta,
                                int nNodes) {
    const int t = blockIdx.x * blockDim.x + threadIdx.x;
    if (t >= nNodes * 16) return;
    const int c = t & 15;
    const float invN = 1.0f / (float)nNodes;
    const float mean = stats[c] * invN;
    const float var  = stats[16 + c] * invN - mean * mean;
    const float v = (x[t] - mean) * rsqrtf(var + BN_EPS) * gamma[c] + beta[c];
    x[t] = fmaxf(v, 0.0f);
}

// ---------------------------------------------------------------------------
// Launch
// ---------------------------------------------------------------------------
extern "C" void kernel_launch(void* const* d_in, const int* in_sizes, int n_in,
                              void* d_out, int out_size, void* d_ws, size_t ws_size,
                              hipStream_t stream) {
    const int N = in_sizes[1] / 128;   // features: (N, 128)
    const int E = in_sizes[0] / 2;     // edge_index: (2, E)

    const int*   row  = (const int*)d_in[0];
    const int*   col  = row + E;
    const float* feat = (const float*)d_in[1];
    const float* W1   = (const float*)d_in[2];   // (256, 16)
    const float* W2   = (const float*)d_in[3];   // (32, 16)
    const float* W3   = (const float*)d_in[4];   // (32, 64)
    const float* b1   = (const float*)d_in[5];
    const float* b2   = (const float*)d_in[6];
    const float* b3   = (const float*)d_in[7];
    const float* g1   = (const float*)d_in[8];
    const float* g2   = (const float*)d_in[9];
    const float* be1  = (const float*)d_in[10];
    const float* be2  = (const float*)d_in[11];

    float* ws      = (float*)d_ws;
    float* inv_deg = ws;                          // N
    float* zbuf    = inv_deg + N;                 // N*64 (max dout)
    float* x1      = zbuf + (size_t)N * 64;       // N*16
    float* x2      = x1 + (size_t)N * 16;         // N*16
    float* stats   = x2 + (size_t)N * 16;         // 32

    float* out = (float*)d_out;                   // N*64

    const int TPB = 256;
    const int tiles = (N + 15) / 16;
    const int denseBlocks = (tiles + 3) / 4;      // 4 waves (tiles) per 128-thread block

    // --- degrees ---
    init_deg_kernel<<<(N + TPB - 1) / TPB, TPB, 0, stream>>>(inv_deg, N);
    count_deg_kernel<<<(E + TPB - 1) / TPB, TPB, 0, stream>>>(row, inv_deg, E);
    invert_deg_kernel<<<(N + TPB - 1) / TPB, TPB, 0, stream>>>(inv_deg, N);

    // --- layer 1: 128 -> 16 ---
    gcn_dense_wmma_kernel<128, 16><<<denseBlocks, 128, 0, stream>>>(
        feat, W1, b1, inv_deg, x1, zbuf, N);
    {
        long long total = (long long)E * 4;       // dout/4 groups per edge
        edge_scatter_kernel<4><<<(int)((total + TPB - 1) / TPB), TPB, 0, stream>>>(
            row, col, inv_deg, zbuf, x1, E);
    }
    zero_stats_kernel<<<1, 32, 0, stream>>>(stats);
    bn_stats_kernel<<<256, TPB, 0, stream>>>(x1, stats, N);
    bn_apply_kernel<<<(N * 16 + TPB - 1) / TPB, TPB, 0, stream>>>(x1, stats, g1, be1, N);

    // --- layer 2: 16 -> 16 ---
    gcn_dense_wmma_kernel<16, 16><<<denseBlocks, 128, 0, stream>>>(
        x1, W2, b2, inv_deg, x2, zbuf, N);
    {
        long long total = (long long)E * 4;
        edge_scatter_kernel<4><<<(int)((total + TPB - 1) / TPB), TPB, 0, stream>>>(
            row, col, inv_deg, zbuf, x2, E);
    }
    zero_stats_kernel<<<1, 32, 0, stream>>>(stats);
    bn_stats_kernel<<<256, TPB, 0, stream>>>(x2, stats, N);
    bn_apply_kernel<<<(N * 16 + TPB - 1) / TPB, TPB, 0, stream>>>(x2, stats, g2, be2, N);

    // --- layer 3: 16 -> 64 (no BN/ReLU), writes d_out ---
    gcn_dense_wmma_kernel<16, 64><<<denseBlocks, 128, 0, stream>>>(
        x2, W3, b3, inv_deg, out, zbuf, N);
    {
        long long total = (long long)E * 16;      // 64/4 groups per edge
        edge_scatter_kernel<6><<<(int)((total + TPB - 1) / TPB), TPB, 0, stream>>>(
            row, col, inv_deg, zbuf, out, E);
    }
}